// MeDiConv_32229434589786
// MI455X (gfx1250) — compile-verified
//
#include <hip/hip_runtime.h>
#include <hip/hip_bf16.h>

typedef __attribute__((ext_vector_type(2))) float v2f;
typedef __attribute__((ext_vector_type(8))) float v8f;

#define BATCH 32
#define IMH   224
#define IMW   224
#define NCH   3
#define NF    64
#define NPIX  (BATCH * IMH * IMW)   // 1,605,632 = 6272 * 256 exactly
#define WTILES (IMW / 16)           // 14 exactly
#define NTILES (BATCH * IMH * WTILES) // 100,352 = 12,544 * 8 exactly

// ---------------------------------------------------------------------------
// Kernel 0: zero the global-sum accumulator cell in workspace
// ---------------------------------------------------------------------------
__global__ void MeDiConv_zero_accum(float* accum) {
    if (blockIdx.x == 0 && threadIdx.x == 0) accum[0] = 0.0f;
}

// ---------------------------------------------------------------------------
// Kernel 1: per-pixel 3x3x3 median (14th smallest of 27, zero-padded SAME),
// write med[] to workspace, and block-reduce + atomicAdd the global sum.
// Median via 27-pass odd-even transposition network: every index is a
// compile-time constant after unrolling -> pure register file, no scratch.
// ---------------------------------------------------------------------------
__global__ __launch_bounds__(256) void MeDiConv_median(
    const float* __restrict__ x, float* __restrict__ med,
    float* __restrict__ accum) {
    const int idx = blockIdx.x * 256 + threadIdx.x;   // pixel id, exact fit
    const int w = idx % IMW;
    const int h = (idx / IMW) % IMH;
    const int b = idx / (IMW * IMH);

    float vals[27];
    int n = 0;
    #pragma unroll
    for (int i = 0; i < 3; ++i) {
        #pragma unroll
        for (int j = 0; j < 3; ++j) {
            const int hh = h + i - 1, ww = w + j - 1;
            const bool ok = (hh >= 0) & (hh < IMH) & (ww >= 0) & (ww < IMW);
            const float* p = x + (((size_t)b * IMH + hh) * IMW + ww) * NCH;
            #pragma unroll
            for (int c = 0; c < 3; ++c) {
                vals[n] = ok ? p[c] : 0.0f;
                ++n;
            }
        }
    }
    // odd-even transposition sort: 27 passes guarantee a fully sorted array
    #pragma unroll
    for (int pass = 0; pass < 27; ++pass) {
        #pragma unroll
        for (int k = (pass & 1); k + 1 < 27; k += 2) {
            const float lo = fminf(vals[k], vals[k + 1]);
            const float hi = fmaxf(vals[k], vals[k + 1]);
            vals[k] = lo;
            vals[k + 1] = hi;
        }
    }
    const float m = vals[13];  // ascending index D - D//2 - 1 = 13
    med[idx] = m;

    // block reduction for global mean
    __shared__ float sh[256];
    sh[threadIdx.x] = m;
    __syncthreads();
    #pragma unroll
    for (int off = 128; off > 0; off >>= 1) {
        if ((int)threadIdx.x < off) sh[threadIdx.x] += sh[threadIdx.x + off];
        __syncthreads();
    }
    if (threadIdx.x == 0) atomicAdd(accum, sh[0]);
}

// ---------------------------------------------------------------------------
// Kernel 2: mean-center + 3x3 conv (1 -> 64 channels) via V_WMMA_F32_16X16X4_F32
// One wave32 handles a tile of 16 consecutive pixels (along W) x all 64 filters.
// GEMM view per tile: [16 x 12] x [12 x 16]x4groups, K padded 9->12 with zeros.
// A layout (16x4 f32): lanes 0-15 hold K=0(v0),K=1(v1); lanes 16-31 K=2,K=3.
// B layout (4x16 f32): VGPR v holds row K = v + 2*half, N = lane&15.
// D layout (16x16 f32): VGPR v holds M = v + 8*half, N = lane&15.
// For both A and B the per-lane K index is t = 4*s + 2*half + c.
// ---------------------------------------------------------------------------
__global__ __launch_bounds__(256) void MeDiConv_conv_wmma(
    const float* __restrict__ med, const float* __restrict__ Wmat,
    const float* __restrict__ accum, float* __restrict__ out) {
    const int wave = threadIdx.x >> 5;
    const int lane = threadIdx.x & 31;
    const int tile = blockIdx.x * 8 + wave;          // exact fit, no guard
    const int wt = tile % WTILES;
    const int h  = (tile / WTILES) % IMH;
    const int b  = tile / (WTILES * IMH);
    const int w0 = wt * 16;
    const int half = lane >> 4;
    const int m    = lane & 15;
    const int wpix = w0 + m;

    const float mean = accum[0] * (1.0f / (float)NPIX);

    // A-operand taps and B-operand weights for this lane.
    // Padding happens AFTER centering in the reference, so OOB taps are 0
    // (not -mean); weights for padded K (t >= 9) are 0.
    float tap[6];
    float wb[6][4];
    #pragma unroll
    for (int s = 0; s < 3; ++s) {
        #pragma unroll
        for (int c = 0; c < 2; ++c) {
            const int t = 4 * s + 2 * half + c;      // K index, lane-dependent
            float a = 0.0f;
            if (t < 9) {
                const int i = t / 3, j = t % 3;
                const int hh = h + i - 1, ww = wpix + j - 1;
                if (hh >= 0 && hh < IMH && ww >= 0 && ww < IMW)
                    a = med[((size_t)b * IMH + hh) * IMW + ww] - mean;
            }
            tap[2 * s + c] = a;
            #pragma unroll
            for (int g = 0; g < 4; ++g) {
                float v = 0.0f;
                if (t < 9) v = Wmat[t * NF + (g * 16 + m)]; // W[i][j][0][n]
                wb[2 * s + c][g] = v;
            }
        }
    }

    const v8f zero = {0.f, 0.f, 0.f, 0.f, 0.f, 0.f, 0.f, 0.f};
    v8f acc[4] = {zero, zero, zero, zero};

    #pragma unroll
    for (int s = 0; s < 3; ++s) {
        v2f a;
        a.x = tap[2 * s];
        a.y = tap[2 * s + 1];
        #pragma unroll
        for (int g = 0; g < 4; ++g) {
            v2f bb;
            bb.x = wb[2 * s][g];
            bb.y = wb[2 * s + 1][g];
            acc[g] = __builtin_amdgcn_wmma_f32_16x16x4_f32(
                /*neg_a=*/false, a, /*neg_b=*/false, bb,
                /*c_mod=*/(short)0, acc[g],
                /*reuse_a=*/false, /*reuse_b=*/false);
        }
    }

    // Coalesced stores: for fixed (g,v), lanes 0-15 and 16-31 each write 64
    // consecutive bytes along the filter dimension.
    const size_t baserow = ((size_t)b * IMH + h) * IMW;
    #pragma unroll
    for (int g = 0; g < 4; ++g) {
        #pragma unroll
        for (int v = 0; v < 8; ++v) {
            const int M = v + 8 * half;              // pixel within tile
            const int n = g * 16 + m;                // filter index
            out[(baserow + (size_t)(w0 + M)) * NF + n] = acc[g][v];
        }
    }
}

// ---------------------------------------------------------------------------
extern "C" void kernel_launch(void* const* d_in, const int* in_sizes, int n_in,
                              void* d_out, int out_size, void* d_ws, size_t ws_size,
                              hipStream_t stream) {
    const float* x    = (const float*)d_in[0];   // [32,224,224,3] f32
    const float* Wmat = (const float*)d_in[1];   // [3,3,1,64] f32
    float* out = (float*)d_out;                  // [32,224,224,64] f32

    float* ws_f  = (float*)d_ws;
    float* accum = ws_f;                         // 1 float (16-float slot)
    float* med   = ws_f + 16;                    // NPIX floats (~6.4 MB)

    MeDiConv_zero_accum<<<1, 1, 0, stream>>>(accum);
    MeDiConv_median<<<NPIX / 256, 256, 0, stream>>>(x, med, accum);
    MeDiConv_conv_wmma<<<NTILES / 8, 256, 0, stream>>>(med, Wmat, accum, out);
}